// Llama4DecoderLayer_9483287789696
// MI455X (gfx1250) — compile-verified
//
#include <hip/hip_runtime.h>
#include <math.h>

// ---------------- model constants (match reference) ----------------
#define D_MODEL 2048
#define NH      16
#define NKV     4
#define DHEAD   128
#define FF      2048
#define NE      8
#define SEQ     512

typedef float v2f __attribute__((ext_vector_type(2)));
typedef float v8f __attribute__((ext_vector_type(8)));

// 16-byte async copy global -> LDS (CDNA5 GLOBAL_LOAD_ASYNC_TO_LDS_B128,
// tracked by ASYNCcnt). VDST = VGPR holding LDS byte address (low 32 bits
// of the generic shared pointer; flat->LDS truncates to addr[31:0]).
__device__ __forceinline__ void async_cp16(const float* gsrc, float* ldst)
{
    unsigned loff = (unsigned)(unsigned long long)(void*)ldst;
    unsigned long long gaddr = (unsigned long long)gsrc;
    asm volatile("global_load_async_to_lds_b128 %0, %1, off"
                 :: "v"(loff), "v"(gaddr) : "memory");
}

// =====================================================================
// Batched GEMM, fp32 via V_WMMA_F32_16X16X4_F32 with LDS double-buffered
// async staging.
//   C[z] = alpha * A[z] @ op(B[z/kvDiv])   (+ C[z] if accumulate, alpha==1)
// A: M x K row-major (lda). B: K x N row-major (ldb), or N x K row-major
// if transB. C: M x N row-major (ldc).
// Block = 256 thr = 8 waves; block tile 64(M) x 64(N); wave tile 16 x 32
// (two 16x16 accumulators sharing one A fragment). K consumed in tiles of
// 32, double-buffered in LDS via GLOBAL_LOAD_ASYNC_TO_LDS_B128 + пipelined
// s_wait_asynccnt. Requires: M,N % 64 == 0, K % 32 == 0, lda/ldb % 4 == 0.
// =====================================================================
__global__ __launch_bounds__(256)
void gemm_f32_wmma(const float* __restrict__ A, const float* __restrict__ B,
                   float* __restrict__ C, int M, int N, int K,
                   int lda, int ldb, int ldc,
                   long sA, long sB, long sC, int kvDiv,
                   float alpha, int transB, int accumulate)
{
    __shared__ float lA[2][64 * 32];     // [row m][k]        (64x32)
    __shared__ float lB[2][64 * 32];     // !transB: [k][n] (32x64); transB: [n][k] (64x32)

    const int tid   = threadIdx.x;
    const int lane  = tid & 31;
    const int wave  = tid >> 5;
    const int waveM = wave >> 1;         // 0..3
    const int waveN = wave & 1;          // 0..1
    const int z     = blockIdx.z;
    const int tileM = blockIdx.y * 64;
    const int tileN = blockIdx.x * 64;

    A += (long)z * sA + (long)tileM * lda;
    B += (long)(z / kvDiv) * sB;
    C += (long)z * sC;

    const int half = lane >> 4;          // 0: K=k,k+1   1: K=k+2,k+3
    const int l    = lane & 15;

    // stage one 64x32 A tile and one 32x64 (or 64x32) B tile: 512 chunks of
    // 16B each per matrix, 2 chunks per thread per matrix -> 4 async ops/wave.
    auto issue = [&](int kt, int buf) {
        const int k0 = kt * 32;
#pragma unroll
        for (int i = 0; i < 2; ++i) {
            const int c  = tid + i * 256;
            const int ra = c >> 3, ca = (c & 7) * 4;      // 64 rows x 8 chunks
            async_cp16(A + (long)ra * lda + k0 + ca, &lA[buf][ra * 32 + ca]);
        }
        if (transB) {
#pragma unroll
            for (int i = 0; i < 2; ++i) {
                const int c  = tid + i * 256;
                const int rb = c >> 3, cb = (c & 7) * 4;  // 64 n-rows x 8 chunks
                async_cp16(B + (long)(tileN + rb) * ldb + k0 + cb,
                           &lB[buf][rb * 32 + cb]);
            }
        } else {
#pragma unroll
            for (int i = 0; i < 2; ++i) {
                const int c  = tid + i * 256;
                const int rb = c >> 4, cb = (c & 15) * 4; // 32 k-rows x 16 chunks
                async_cp16(B + (long)(k0 + rb) * ldb + tileN + cb,
                           &lB[buf][rb * 64 + cb]);
            }
        }
    };

    v8f c0 = {0.f, 0.f, 0.f, 0.f, 0.f, 0.f, 0.f, 0.f};
    v8f c1 = {0.f, 0.f, 0.f, 0.f, 0.f, 0.f, 0.f, 0.f};
    const int rowC = tileM + waveM * 16;
    const int colC = tileN + waveN * 32;
    if (accumulate) {
#pragma unroll
        for (int r = 0; r < 8; ++r) {
            const long row = (long)(rowC + r + 8 * half);
            c0[r] = C[row * ldc + colC + l];
            c1[r] = C[row * ldc + colC + 16 + l];
        }
    }

    const int KT = K >> 5;
    issue(0, 0);
    for (int kt = 0; kt < KT; ++kt) {
        const int buf = kt & 1;
        if (kt + 1 < KT) {
            issue(kt + 1, buf ^ 1);
            asm volatile("s_wait_asynccnt 0x4" ::: "memory"); // prev tile done
        } else {
            asm volatile("s_wait_asynccnt 0x0" ::: "memory");
        }
        __syncthreads();                                      // tile visible to all waves

        const float* As = &lA[buf][(waveM * 16 + l) * 32 + 2 * half];
        if (transB) {
            const float* B0 = &lB[buf][(waveN * 32 + l) * 32 + 2 * half];
            const float* B1 = B0 + 16 * 32;
#pragma unroll
            for (int kk = 0; kk < 32; kk += 4) {
                v2f a  = *(const v2f*)(As + kk);
                v2f b0 = *(const v2f*)(B0 + kk);
                v2f b1 = *(const v2f*)(B1 + kk);
                c0 = __builtin_amdgcn_wmma_f32_16x16x4_f32(
                    false, a, false, b0, (short)0, c0, false, false);
                c1 = __builtin_amdgcn_wmma_f32_16x16x4_f32(
                    false, a, false, b1, (short)0, c1, false, false);
            }
        } else {
            const float* Bb = &lB[buf][waveN * 32 + l];
#pragma unroll
            for (int kk = 0; kk < 32; kk += 4) {
                const int kr = kk + 2 * half;
                v2f a = *(const v2f*)(As + kk);
                v2f b0, b1;
                b0.x = Bb[kr * 64];       b0.y = Bb[(kr + 1) * 64];
                b1.x = Bb[kr * 64 + 16];  b1.y = Bb[(kr + 1) * 64 + 16];
                c0 = __builtin_amdgcn_wmma_f32_16x16x4_f32(
                    false, a, false, b0, (short)0, c0, false, false);
                c1 = __builtin_amdgcn_wmma_f32_16x16x4_f32(
                    false, a, false, b1, (short)0, c1, false, false);
            }
        }
        __syncthreads();   // all waves done with buf before it is re-filled
    }

#pragma unroll
    for (int r = 0; r < 8; ++r) {
        const long row = (long)(rowC + r + 8 * half);
        const float v0 = accumulate ? c0[r] : alpha * c0[r];
        const float v1 = accumulate ? c1[r] : alpha * c1[r];
        C[row * ldc + colC + l]      = v0;
        C[row * ldc + colC + 16 + l] = v1;
    }
}

// ---------------- RMSNorm: one block per token ----------------
__global__ __launch_bounds__(256)
void rmsnorm_kernel(const float* __restrict__ in, const float* __restrict__ w,
                    float* __restrict__ out)
{
    const int t = blockIdx.x;
    const float* row = in + (long)t * D_MODEL;
    float s = 0.f;
    for (int i = threadIdx.x; i < D_MODEL; i += 256) { float v = row[i]; s += v * v; }
    __shared__ float red[256];
    red[threadIdx.x] = s; __syncthreads();
    for (int o = 128; o > 0; o >>= 1) {
        if (threadIdx.x < o) red[threadIdx.x] += red[threadIdx.x + o];
        __syncthreads();
    }
    const float rs = rsqrtf(red[0] / (float)D_MODEL + 1e-5f);
    for (int i = threadIdx.x; i < D_MODEL; i += 256)
        out[(long)t * D_MODEL + i] = row[i] * rs * w[i];
}

// ---------------- RoPE in-place on (heads, SEQ, DHEAD), 64 thr/pair ----
__global__ __launch_bounds__(64)
void rope_kernel(float* __restrict__ x, const int* __restrict__ pos_ids)
{
    const int hs = blockIdx.x;
    const int h = hs / SEQ, t = hs % SEQ;
    const int i = threadIdx.x;               // 0..63  (pair index)
    float* p = x + ((long)h * SEQ + t) * DHEAD;
    const float pos = (float)pos_ids[t];
    const float inv = powf(500000.0f, -(2.0f * (float)i) / 128.0f);
    const float f = pos * inv;
    const float c = cosf(f), s = sinf(f);
    const float x1 = p[i], x2 = p[i + 64];
    p[i]      = x1 * c - x2 * s;
    p[i + 64] = x2 * c + x1 * s;
}

// ---------------- causal softmax over scores (NH, SEQ, SEQ) ----------
__global__ __launch_bounds__(256)
void softmax_kernel(float* __restrict__ sc, const int* __restrict__ amask)
{
    const int h = blockIdx.y, q = blockIdx.x;
    float* row = sc + ((long)h * SEQ + q) * SEQ;
    __shared__ float red[256];
    float m = -1e30f;
    for (int j = threadIdx.x; j < SEQ; j += 256) {
        float v = (j <= q && amask[j] > 0) ? row[j] : -1e9f;
        m = fmaxf(m, v);
    }
    red[threadIdx.x] = m; __syncthreads();
    for (int o = 128; o > 0; o >>= 1) {
        if (threadIdx.x < o) red[threadIdx.x] = fmaxf(red[threadIdx.x], red[threadIdx.x + o]);
        __syncthreads();
    }
    m = red[0]; __syncthreads();
    float s = 0.f;
    for (int j = threadIdx.x; j < SEQ; j += 256) {
        float v = (j <= q && amask[j] > 0) ? row[j] : -1e9f;
        float e = __expf(v - m);
        row[j] = e;
        s += e;
    }
    red[threadIdx.x] = s; __syncthreads();
    for (int o = 128; o > 0; o >>= 1) {
        if (threadIdx.x < o) red[threadIdx.x] += red[threadIdx.x + o];
        __syncthreads();
    }
    const float invs = 1.0f / red[0];
    for (int j = threadIdx.x; j < SEQ; j += 256) row[j] *= invs;
}

// ---------------- residual add, dual write ----------------
__global__ __launch_bounds__(256)
void add2_kernel(const float* __restrict__ a, const float* __restrict__ b,
                 float* __restrict__ o1, float* __restrict__ o2)
{
    const long i = (long)blockIdx.x * 256 + threadIdx.x;
    const float v = a[i] + b[i];
    o1[i] = v; o2[i] = v;
}

// ---------------- router: logits -> top-1 idx + sigmoid score ---------
__global__ __launch_bounds__(256)
void router_kernel(const float* __restrict__ x2, const float* __restrict__ rw,
                   float* __restrict__ scale, int* __restrict__ idx)
{
    const int t = blockIdx.x;
    const int e = threadIdx.x >> 5;      // wave e computes expert e logit
    const int lane = threadIdx.x & 31;
    const float* row = x2 + (long)t * D_MODEL;
    float s = 0.f;
    for (int k = lane; k < D_MODEL; k += 32) s += row[k] * rw[(long)k * NE + e];
    for (int o = 16; o > 0; o >>= 1) s += __shfl_down(s, o);
    __shared__ float logit[NE];
    if (lane == 0) logit[e] = s;
    __syncthreads();
    if (threadIdx.x == 0) {
        int best = 0; float bv = logit[0];
        for (int j = 1; j < NE; ++j) if (logit[j] > bv) { bv = logit[j]; best = j; }
        scale[t] = 1.0f / (1.0f + __expf(-bv));
        idx[t]   = best;
    }
}

// ---------------- masked scaled SiLU gating for routed experts --------
__global__ __launch_bounds__(256)
void moe_act_kernel(const float* __restrict__ g, const float* __restrict__ u,
                    const float* __restrict__ scale, const int* __restrict__ idx,
                    float* __restrict__ out, int e)
{
    const long i = (long)blockIdx.x * 256 + threadIdx.x;   // over SEQ*FF
    const int t = (int)(i / FF);
    float r = 0.f;
    if (idx[t] == e) {
        const float sc = scale[t];
        const float gv = sc * g[i], uv = sc * u[i];
        r = (gv / (1.0f + __expf(-gv))) * uv;
    }
    out[i] = r;
}

// ---------------- shared-expert SiLU gating ----------------
__global__ __launch_bounds__(256)
void shared_act_kernel(const float* __restrict__ g, const float* __restrict__ u,
                       float* __restrict__ out)
{
    const long i = (long)blockIdx.x * 256 + threadIdx.x;
    const float gv = g[i];
    out[i] = (gv / (1.0f + __expf(-gv))) * u[i];
}

// =====================================================================
static inline void gemm(hipStream_t st, const float* A, const float* B, float* C,
                        int M, int N, int K, int lda, int ldb, int ldc,
                        long sA, long sB, long sC, int kvDiv, int batch,
                        float alpha, int transB, int acc)
{
    dim3 grid(N / 64, M / 64, batch), block(256);
    gemm_f32_wmma<<<grid, block, 0, st>>>(A, B, C, M, N, K, lda, ldb, ldc,
                                          sA, sB, sC, kvDiv, alpha, transB, acc);
}

extern "C" void kernel_launch(void* const* d_in, const int* in_sizes, int n_in,
                              void* d_out, int out_size, void* d_ws, size_t ws_size,
                              hipStream_t stream)
{
    const float* hidden = (const float*)d_in[0];
    const int*   amask  = (const int*)d_in[1];
    const int*   pos    = (const int*)d_in[2];
    const float* anw    = (const float*)d_in[3];
    const float* wq     = (const float*)d_in[4];
    const float* wk     = (const float*)d_in[5];
    const float* wv     = (const float*)d_in[6];
    const float* wo     = (const float*)d_in[7];
    const float* fnw    = (const float*)d_in[8];
    const float* rw     = (const float*)d_in[9];
    const float* wgate  = (const float*)d_in[10];
    const float* wup    = (const float*)d_in[11];
    const float* wdown  = (const float*)d_in[12];
    const float* wsg    = (const float*)d_in[13];
    const float* wsu    = (const float*)d_in[14];
    const float* wsd    = (const float*)d_in[15];
    float* out = (float*)d_out;

    // workspace layout (floats)
    float* ws = (float*)d_ws;
    const long SZ_SD  = (long)SEQ * D_MODEL;          // 1048576
    const long SZ_KV  = (long)NKV * SEQ * DHEAD;      // 262144
    const long SZ_SC  = (long)NH * SEQ * SEQ;         // 4194304
    float* X   = ws;                                  // normed input
    float* Q   = X   + SZ_SD;                         // (NH,SEQ,DHEAD)
    float* KB  = Q   + SZ_SD;                         // (NKV,SEQ,DHEAD)
    float* VB  = KB  + SZ_KV;
    float* SC  = VB  + SZ_KV;                         // scores / wo tmp
    float* ATT = SC  + SZ_SC;                         // (SEQ, NH*DHEAD)
    float* HR  = ATT + SZ_SD;                         // residual h
    float* X2  = HR  + SZ_SD;                         // normed h
    float* GB  = X2  + SZ_SD;                         // gate tmp
    float* UB  = GB  + SZ_SD;                         // up tmp
    float* HB  = UB  + SZ_SD;                         // gated tmp
    float* SCALE = HB + SZ_SD;                        // 512 floats
    int*   IDX   = (int*)(SCALE + SEQ);               // 512 ints

    const dim3 b256(256);
    const int  nSD = (int)(SZ_SD / 256);              // 4096 blocks

    // ---- attention block ----
    rmsnorm_kernel<<<SEQ, b256, 0, stream>>>(hidden, anw, X);

    // per-head Q/K/V projections (batched over heads, B cols offset by 128)
    gemm(stream, X, wq, Q,  SEQ, DHEAD, D_MODEL, D_MODEL, NH * DHEAD,  DHEAD,
         0, DHEAD, (long)SEQ * DHEAD, 1, NH,  1.f, 0, 0);
    gemm(stream, X, wk, KB, SEQ, DHEAD, D_MODEL, D_MODEL, NKV * DHEAD, DHEAD,
         0, DHEAD, (long)SEQ * DHEAD, 1, NKV, 1.f, 0, 0);
    gemm(stream, X, wv, VB, SEQ, DHEAD, D_MODEL, D_MODEL, NKV * DHEAD, DHEAD,
         0, DHEAD, (long)SEQ * DHEAD, 1, NKV, 1.f, 0, 0);

    rope_kernel<<<NH * SEQ,  dim3(64), 0, stream>>>(Q,  pos);
    rope_kernel<<<NKV * SEQ, dim3(64), 0, stream>>>(KB, pos);

    // scores = Q @ K^T / sqrt(DH)   (GQA: kv head = h/4)
    gemm(stream, Q, KB, SC, SEQ, SEQ, DHEAD, DHEAD, DHEAD, SEQ,
         (long)SEQ * DHEAD, (long)SEQ * DHEAD, (long)SEQ * SEQ, NKV, NH,
         0.088388347648318447f, 1, 0);

    softmax_kernel<<<dim3(SEQ, NH), b256, 0, stream>>>(SC, amask);

    // attn = P @ V  -> (SEQ, NH*DHEAD) head-interleaved columns
    gemm(stream, SC, VB, ATT, SEQ, DHEAD, SEQ, SEQ, DHEAD, NH * DHEAD,
         (long)SEQ * SEQ, (long)SEQ * DHEAD, DHEAD, NKV, NH, 1.f, 0, 0);

    // wo projection into SC (reused as tmp), then residual h = hidden + attn
    gemm(stream, ATT, wo, SC, SEQ, D_MODEL, NH * DHEAD,
         NH * DHEAD, D_MODEL, D_MODEL, 0, 0, 0, 1, 1, 1.f, 0, 0);
    add2_kernel<<<nSD, b256, 0, stream>>>(hidden, SC, HR, out);   // out starts as h

    // ---- MoE block ----
    rmsnorm_kernel<<<SEQ, b256, 0, stream>>>(HR, fnw, X2);
    router_kernel<<<SEQ, b256, 0, stream>>>(X2, rw, SCALE, IDX);

    // shared expert: out += (silu(x2@wsg) * (x2@wsu)) @ wsd
    gemm(stream, X2, wsg, GB, SEQ, FF, D_MODEL, D_MODEL, FF, FF, 0, 0, 0, 1, 1, 1.f, 0, 0);
    gemm(stream, X2, wsu, UB, SEQ, FF, D_MODEL, D_MODEL, FF, FF, 0, 0, 0, 1, 1, 1.f, 0, 0);
    shared_act_kernel<<<nSD, b256, 0, stream>>>(GB, UB, HB);
    gemm(stream, HB, wsd, out, SEQ, D_MODEL, FF, FF, D_MODEL, D_MODEL,
         0, 0, 0, 1, 1, 1.f, 0, 1);

    // routed experts (top-1; masked-scaled SiLU before down proj, accumulate)
    for (int e = 0; e < NE; ++e) {
        const float* ge = wgate + (long)e * D_MODEL * FF;
        const float* ue = wup   + (long)e * D_MODEL * FF;
        const float* de = wdown + (long)e * FF * D_MODEL;
        gemm(stream, X2, ge, GB, SEQ, FF, D_MODEL, D_MODEL, FF, FF, 0, 0, 0, 1, 1, 1.f, 0, 0);
        gemm(stream, X2, ue, UB, SEQ, FF, D_MODEL, D_MODEL, FF, FF, 0, 0, 0, 1, 1, 1.f, 0, 0);
        moe_act_kernel<<<nSD, b256, 0, stream>>>(GB, UB, SCALE, IDX, HB, e);
        gemm(stream, HB, de, out, SEQ, D_MODEL, FF, FF, D_MODEL, D_MODEL,
             0, 0, 0, 1, 1, 1.f, 0, 1);
    }
    (void)in_sizes; (void)n_in; (void)out_size; (void)ws_size;
}